// PWPNNFcn_53171695125376
// MI455X (gfx1250) — compile-verified
//
#include <hip/hip_runtime.h>
#include <hip/hip_bf16.h>

typedef __attribute__((ext_vector_type(2))) float v2f;
typedef __attribute__((ext_vector_type(8))) float v8f;

constexpr int N_SMPS = 4096;
constexpr int D_IN   = 64;
constexpr int D_OUT  = 128;
constexpr int N_FCNS = 4096;
constexpr int K_NN   = 4;
constexpr float FLT_BIG = 3.402823466e+38f;

// Insert (d,id) into ascending sorted 4-list (s0<=s1<=s2<=s3). Strict '<'
// keeps earliest index on ties (matches jax top_k tie-break).
#define INSERT4(d, id)                                            \
  {                                                               \
    bool c3 = (d) < s3, c2 = (d) < s2, c1 = (d) < s1, c0 = (d) < s0; \
    s3 = c3 ? (c2 ? s2 : (d)) : s3;  i3 = c3 ? (c2 ? i2 : (id)) : i3; \
    s2 = c2 ? (c1 ? s1 : (d)) : s2;  i2 = c2 ? (c1 ? i1 : (id)) : i2; \
    s1 = c1 ? (c0 ? s0 : (d)) : s1;  i1 = c1 ? (c0 ? i0 : (id)) : i1; \
    s0 = c0 ? (d) : s0;              i0 = c0 ? (id) : i0;             \
  }

// ---------------------------------------------------------------------------
// K0: zero the output accumulator and the per-function counters.
// ---------------------------------------------------------------------------
__global__ __launch_bounds__(256) void init_kernel(float* __restrict__ out,
                                                   int* __restrict__ cnt) {
  int t = blockIdx.x * blockDim.x + threadIdx.x;
  if (t < N_SMPS * D_OUT) out[t] = 0.0f;
  if (t < N_FCNS) cnt[t] = 0;
}

// ---------------------------------------------------------------------------
// K1: fused distance GEMM (x @ ctrs^T via V_WMMA_F32_16X16X4_F32) + top-4.
// One wave owns a 16-sample row tile and scans all 4096 centers in 16-wide
// tiles (16 chained K=4 WMMAs per tile). ||c||^2 is accumulated from the
// same registers used for the B operand, halves combined with shfl_xor(16).
// Ranking key: ||c||^2 - 2*(x.c)  (||x||^2 is row-constant).
// 4 waves / block, 64 blocks -> 4096 samples.
// ---------------------------------------------------------------------------
__global__ __launch_bounds__(128) void knn_kernel(const float* __restrict__ x,
                                                  const float* __restrict__ ctrs,
                                                  int* __restrict__ idx) {
  __shared__ float Glds[4][16 * 16];
  __shared__ float CNlds[4][16];
  const int wave = threadIdx.x >> 5;
  const int lane = threadIdx.x & 31;
  const int lmod = lane & 15;
  const int lhi  = lane >> 4;
  const int mrow = (blockIdx.x * 4 + wave) * 16 + lmod;  // sample row
  float* G  = Glds[wave];
  float* CN = CNlds[wave];

  // Preload A operand registers: A is 16x4 per WMMA; lane l holds row l%16,
  // K = 4*j + 2*(l/16) + {0,1} in VGPR pair j.
  v2f a[16];
#pragma unroll
  for (int j = 0; j < 16; ++j)
    a[j] = *(const v2f*)(x + mrow * D_IN + 4 * j + 2 * lhi);

  float s0 = FLT_BIG, s1 = FLT_BIG, s2 = FLT_BIG, s3 = FLT_BIG;
  int i0 = 0, i1 = 0, i2 = 0, i3 = 0;

  for (int ct = 0; ct < N_FCNS / 16; ++ct) {
    v8f acc = {0.f, 0.f, 0.f, 0.f, 0.f, 0.f, 0.f, 0.f};
    float cn = 0.0f;
#pragma unroll
    for (int j = 0; j < 16; ++j) {
      // B operand (4x16 tile of ctrs^T): lane l holds column N=l%16,
      // K = 4*j + 2*(l/16) + {0,1} (mirror of the A layout).
      v2f b = *(const v2f*)(ctrs + (ct * 16 + lmod) * D_IN + 4 * j + 2 * lhi);
      cn += b.x * b.x + b.y * b.y;
      acc = __builtin_amdgcn_wmma_f32_16x16x4_f32(
          false, a[j], false, b, (short)0, acc, false, false);
    }
    // lane l and l+16 together cover all 64 K for center l%16
    cn += __shfl_xor(cn, 16, 32);
    if (lhi == 0) CN[lmod] = cn;
    // C layout: VGPR r of lane l = G[M = r + 8*(l/16)][N = l%16]
#pragma unroll
    for (int r = 0; r < 8; ++r) G[(r + 8 * lhi) * 16 + lmod] = acc[r];
    __builtin_amdgcn_wave_barrier();  // wave-private LDS; just stop reordering
    // Lane l scans candidates n in [8*lhi, 8*lhi+8) for its sample row l%16
#pragma unroll
    for (int q = 0; q < 8; ++q) {
      int n = 8 * lhi + q;
      float d = CN[n] - 2.0f * G[lmod * 16 + n];
      INSERT4(d, ct * 16 + n);
    }
    __builtin_amdgcn_wave_barrier();
  }

  // Merge the two half-lane top-4 lists for the same sample row.
  float t0 = __shfl_xor(s0, 16, 32), t1 = __shfl_xor(s1, 16, 32);
  float t2 = __shfl_xor(s2, 16, 32), t3 = __shfl_xor(s3, 16, 32);
  int j0 = __shfl_xor(i0, 16, 32), j1 = __shfl_xor(i1, 16, 32);
  int j2 = __shfl_xor(i2, 16, 32), j3 = __shfl_xor(i3, 16, 32);
  INSERT4(t0, j0);
  INSERT4(t1, j1);
  INSERT4(t2, j2);
  INSERT4(t3, j3);
  if (lhi == 0) {
    idx[mrow * K_NN + 0] = i0;
    idx[mrow * K_NN + 1] = i1;
    idx[mrow * K_NN + 2] = i2;
    idx[mrow * K_NN + 3] = i3;
  }
}

// ---------------------------------------------------------------------------
// K2: histogram of selected functions.
// ---------------------------------------------------------------------------
__global__ __launch_bounds__(256) void count_kernel(const int* __restrict__ idx,
                                                    int* __restrict__ cnt) {
  int t = blockIdx.x * blockDim.x + threadIdx.x;
  if (t < N_SMPS * K_NN) atomicAdd(&cnt[idx[t]], 1);
}

// ---------------------------------------------------------------------------
// K3: exclusive prefix scan over 4096 counts (single block, 256 threads x 16).
// Writes base[] and an identical cursor[] for the scatter pass.
// ---------------------------------------------------------------------------
__global__ __launch_bounds__(256) void scan_kernel(const int* __restrict__ cnt,
                                                   int* __restrict__ base,
                                                   int* __restrict__ cursor) {
  __shared__ int sums[256];
  int t = threadIdx.x;
  int local[16];
  int s = 0;
#pragma unroll
  for (int q = 0; q < 16; ++q) {
    local[q] = s;
    s += cnt[t * 16 + q];
  }
  sums[t] = s;
  __syncthreads();
  for (int off = 1; off < 256; off <<= 1) {
    int v = (t >= off) ? sums[t - off] : 0;
    __syncthreads();
    sums[t] += v;
    __syncthreads();
  }
  int prefix = (t == 0) ? 0 : sums[t - 1];
#pragma unroll
  for (int q = 0; q < 16; ++q) {
    int b = prefix + local[q];
    base[t * 16 + q] = b;
    cursor[t * 16 + q] = b;
  }
}

// ---------------------------------------------------------------------------
// K4: scatter sample ids into per-function contiguous lists.
// ---------------------------------------------------------------------------
__global__ __launch_bounds__(256) void scatter_kernel(const int* __restrict__ idx,
                                                      int* __restrict__ cursor,
                                                      int* __restrict__ pairs) {
  int t = blockIdx.x * blockDim.x + threadIdx.x;
  if (t < N_SMPS * K_NN) {
    int f = idx[t];
    int pos = atomicAdd(&cursor[f], 1);
    pairs[pos] = t >> 2;  // sample id
  }
}

// ---------------------------------------------------------------------------
// K5: inverted apply. One workgroup per function f; each wts[f] (64 KB) is
// streamed from HBM exactly once (268 MB total instead of 1.07 GB gathered).
// A tile (16 samples x K=128 where K = [order0 | order1]) is built in LDS as
// diff = x^p - c^p; B = wts[f] viewed as 128x128 stays register-resident per
// wave (64 VGPRs). out accumulated with global_atomic_add_f32; offsets[f]
// folded into the atomic add per valid row.
// ---------------------------------------------------------------------------
__global__ __launch_bounds__(256) void apply_kernel(
    const float* __restrict__ x, const float* __restrict__ ctrs,
    const float* __restrict__ wts, const float* __restrict__ offsets,
    const int* __restrict__ cnt, const int* __restrict__ base,
    const int* __restrict__ pairs, float* __restrict__ out) {
  __shared__ float A[16 * 128];
  __shared__ int ns[16];

  const int f = blockIdx.x;
  const int cf = cnt[f];
  if (cf == 0) return;  // uniform across the block
  const int bf = base[f];

  const int lane = threadIdx.x & 31;
  const int wave = threadIdx.x >> 5;  // 8 waves = 8 N-tiles of 16 columns
  const int lmod = lane & 15;
  const int lhi  = lane >> 4;
  const int ncol = wave * 16 + lmod;

  const float* wf = wts + (size_t)f * (2 * D_IN * D_OUT);

  // B register-resident: B[k][n] = wts[f, k/64, k%64, n], k = 0..127.
  // Lane l holds column N=ncol, K = 4*kk + 2*(l/16) + {0,1} in pair kk.
  v2f bb[32];
#pragma unroll
  for (int kk = 0; kk < 32; ++kk) {
    int k0 = 4 * kk + 2 * lhi;
    v2f b;
    b.x = wf[(size_t)k0 * D_OUT + ncol];
    b.y = wf[(size_t)(k0 + 1) * D_OUT + ncol];
    bb[kk] = b;
  }
  const float offv = offsets[f * D_OUT + ncol];

  const int nchunk = (cf + 15) >> 4;
  for (int ch = 0; ch < nchunk; ++ch) {
    const int rem = cf - ch * 16;
    {
      // Cooperative A build: thread t -> sample slot s = t/16, 4 inputs each.
      int s = threadIdx.x >> 4;
      int ii = (threadIdx.x & 15) * 4;
      int n = (s < rem) ? pairs[bf + ch * 16 + s] : -1;
      if (ii == 0) ns[s] = n;
      if (n >= 0) {
#pragma unroll
        for (int q = 0; q < 4; ++q) {
          float xv = x[n * D_IN + ii + q];
          float cv = ctrs[f * D_IN + ii + q];
          A[s * 128 + ii + q] = xv - cv;               // order 1
          A[s * 128 + 64 + ii + q] = xv * xv - cv * cv;  // order 2
        }
      } else {
#pragma unroll
        for (int q = 0; q < 4; ++q) {
          A[s * 128 + ii + q] = 0.0f;
          A[s * 128 + 64 + ii + q] = 0.0f;
        }
      }
    }
    __syncthreads();

    v8f acc = {0.f, 0.f, 0.f, 0.f, 0.f, 0.f, 0.f, 0.f};
#pragma unroll
    for (int kk = 0; kk < 32; ++kk) {
      v2f av = *(const v2f*)(&A[lmod * 128 + 4 * kk + 2 * lhi]);
      acc = __builtin_amdgcn_wmma_f32_16x16x4_f32(
          false, av, false, bb[kk], (short)0, acc, false, false);
    }

#pragma unroll
    for (int r = 0; r < 8; ++r) {
      int m = r + 8 * lhi;  // output row of this accumulator VGPR
      if (m < rem) {
        int n = ns[m];
        unsafeAtomicAdd(&out[n * D_OUT + ncol], acc[r] + offv);
      }
    }
    __syncthreads();
  }
}

// ---------------------------------------------------------------------------
extern "C" void kernel_launch(void* const* d_in, const int* in_sizes, int n_in,
                              void* d_out, int out_size, void* d_ws, size_t ws_size,
                              hipStream_t stream) {
  const float* x       = (const float*)d_in[0];
  const float* ctrs    = (const float*)d_in[1];
  const float* wts     = (const float*)d_in[2];
  const float* offsets = (const float*)d_in[3];
  float* out = (float*)d_out;

  // Workspace layout (ints): idx[16384] cnt[4096] base[4096] cursor[4096]
  // pairs[16384]  -> 176 KB.
  int* idx    = (int*)d_ws;
  int* cnt    = idx + N_SMPS * K_NN;
  int* base   = cnt + N_FCNS;
  int* cursor = base + N_FCNS;
  int* pairs  = cursor + N_FCNS;

  init_kernel<<<(N_SMPS * D_OUT + 255) / 256, 256, 0, stream>>>(out, cnt);
  knn_kernel<<<N_SMPS / 64, 128, 0, stream>>>(x, ctrs, idx);
  count_kernel<<<(N_SMPS * K_NN + 255) / 256, 256, 0, stream>>>(idx, cnt);
  scan_kernel<<<1, 256, 0, stream>>>(cnt, base, cursor);
  scatter_kernel<<<(N_SMPS * K_NN + 255) / 256, 256, 0, stream>>>(idx, cursor, pairs);
  apply_kernel<<<N_FCNS, 256, 0, stream>>>(x, ctrs, wts, offsets, cnt, base, pairs, out);
}